// GCN_18674517803330
// MI455X (gfx1250) — compile-verified
//
#include <hip/hip_runtime.h>
#include <hip/hip_bf16.h>

#define HIDDEN 128
#define N_GRAPHS 500

typedef __attribute__((ext_vector_type(2))) float v2f;
typedef __attribute__((ext_vector_type(8))) float v8f;

// ---------------- utility ----------------
__global__ void zero_f32(float* __restrict__ p, int n) {
    int t = blockIdx.x * blockDim.x + threadIdx.x;
    if (t < n) p[t] = 0.0f;
}

// deg[dst] += 1 for every edge
__global__ void deg_kernel(const int* __restrict__ dst, float* __restrict__ deg, int E) {
    int t = blockIdx.x * blockDim.x + threadIdx.x;
    if (t < E) atomicAdd(&deg[dst[t]], 1.0f);
}

// dinv = rsqrt(deg + 1)  (in place)
__global__ void dinv_kernel(float* __restrict__ deg, int n) {
    int t = blockIdx.x * blockDim.x + threadIdx.x;
    if (t < n) deg[t] = rsqrtf(deg[t] + 1.0f);
}

// ---------------- layer 1: outer product [N,1]x[1,128], fused self-loop init
__global__ void layer1_kernel(const float* __restrict__ x, const float* __restrict__ W1,
                              const float* __restrict__ dinv,
                              float* __restrict__ pre, float* __restrict__ agg, int N) {
    int t = blockIdx.x * blockDim.x + threadIdx.x;           // over N*128
    int i = t >> 7;
    int j = t & (HIDDEN - 1);
    if (i < N) {
        float v = x[i] * W1[j];
        float d = dinv[i];
        pre[t] = v;
        agg[t] = v * d * d;
    }
}

// ---------------- fp32 WMMA GEMM: Out = A[N,128] @ W[128,128], fused self-loop init
// grid.x = N/16 blocks, 256 threads = 8 waves; wave w -> output cols [16w,16w+16)
__global__ __launch_bounds__(256) void gemm_wmma_kernel(
        const float* __restrict__ A, const float* __restrict__ W,
        const float* __restrict__ dinv,
        float* __restrict__ pre, float* __restrict__ agg) {
    const int row0 = blockIdx.x * 16;
    const int wave = threadIdx.x >> 5;          // 0..7
    const int lane = threadIdx.x & 31;
    const int col0 = wave * 16;
    const int half = lane >> 4;                 // 0 or 1
    const int l15  = lane & 15;

    const float* __restrict__ Arow = A + (size_t)(row0 + l15) * HIDDEN;

    v8f c = {};
    #pragma unroll
    for (int k0 = 0; k0 < HIDDEN; k0 += 4) {
        const int ka = k0 + half * 2;           // this lane's two K values: ka, ka+1
        v2f a, b;
        a.x = Arow[ka];
        a.y = Arow[ka + 1];
        b.x = W[(size_t)ka * HIDDEN + col0 + l15];
        b.y = W[(size_t)(ka + 1) * HIDDEN + col0 + l15];
        // D = A(16x4 f32) * B(4x16 f32) + C  -- exact fp32 matrix op
        c = __builtin_amdgcn_wmma_f32_16x16x4_f32(
                /*neg_a=*/false, a, /*neg_b=*/false, b,
                /*c_mod=*/(short)0, c, /*reuse_a=*/false, /*reuse_b=*/false);
    }

    #pragma unroll
    for (int v = 0; v < 8; ++v) {
        const int r = row0 + v + 8 * half;      // C layout: M = vgpr + 8*(lane>>4)
        const float val = c[v];
        const float d = dinv[r];
        pre[(size_t)r * HIDDEN + col0 + l15] = val;
        agg[(size_t)r * HIDDEN + col0 + l15] = val * d * d;
    }
}

// ---------------- edge scatter: agg[dst] += pre[src] * dinv[src]*dinv[dst]
// one 32-lane wave per edge; each lane owns 4 consecutive columns (float4)
__global__ __launch_bounds__(256) void scatter_kernel(
        const float* __restrict__ pre, const int* __restrict__ src,
        const int* __restrict__ dst, const float* __restrict__ dinv,
        float* __restrict__ agg, int E) {
    long long t = (long long)blockIdx.x * blockDim.x + threadIdx.x;
    int e = (int)(t >> 5);
    int lane = (int)(t & 31);
    if (e >= E) return;
    int s = src[e];
    int d = dst[e];
    float norm = dinv[s] * dinv[d];
    const float4 h4 = reinterpret_cast<const float4*>(pre + (size_t)s * HIDDEN)[lane];
    float* base = agg + (size_t)d * HIDDEN + lane * 4;
    atomicAdd(base + 0, h4.x * norm);
    atomicAdd(base + 1, h4.y * norm);
    atomicAdd(base + 2, h4.z * norm);
    atomicAdd(base + 3, h4.w * norm);
}

// ---------------- finalize: agg = (relu?)(agg + b)
__global__ void finalize_kernel(float* __restrict__ agg, const float* __restrict__ b,
                                int N, int relu) {
    int t = blockIdx.x * blockDim.x + threadIdx.x;
    int i = t >> 7;
    int j = t & (HIDDEN - 1);
    if (i < N) {
        float v = agg[t] + b[j];
        if (relu) v = fmaxf(v, 0.0f);
        agg[t] = v;
    }
}

// ---------------- mean pool: sums[batch[i]] += h[i]; counts[batch[i]] += 1
__global__ __launch_bounds__(256) void pool_kernel(
        const float* __restrict__ h, const int* __restrict__ batch,
        float* __restrict__ sums, float* __restrict__ counts, int N) {
    long long t = (long long)blockIdx.x * blockDim.x + threadIdx.x;
    int i = (int)(t >> 5);
    int lane = (int)(t & 31);
    if (i >= N) return;
    int g = batch[i];
    const float4 h4 = reinterpret_cast<const float4*>(h + (size_t)i * HIDDEN)[lane];
    float* base = sums + (size_t)g * HIDDEN + lane * 4;
    atomicAdd(base + 0, h4.x);
    atomicAdd(base + 1, h4.y);
    atomicAdd(base + 2, h4.z);
    atomicAdd(base + 3, h4.w);
    if (lane == 0) atomicAdd(&counts[g], 1.0f);
}

// ---------------- head: out[g][c] = mean_g . Wlin[:,c] + blin[c]
__global__ void head_kernel(const float* __restrict__ sums, const float* __restrict__ counts,
                            const float* __restrict__ Wlin, const float* __restrict__ blin,
                            float* __restrict__ out, int nout) {
    int t = blockIdx.x * blockDim.x + threadIdx.x;   // over N_GRAPHS*2
    if (t >= nout) return;
    int g = t >> 1;
    int c = t & 1;
    float inv = 1.0f / fmaxf(counts[g], 1.0f);
    float acc = 0.0f;
    #pragma unroll 8
    for (int j = 0; j < HIDDEN; ++j)
        acc += sums[(size_t)g * HIDDEN + j] * inv * Wlin[j * 2 + c];
    out[t] = acc + blin[c];
}

extern "C" void kernel_launch(void* const* d_in, const int* in_sizes, int n_in,
                              void* d_out, int out_size, void* d_ws, size_t ws_size,
                              hipStream_t stream) {
    const float* x    = (const float*)d_in[0];
    const float* W1   = (const float*)d_in[1];
    const float* b1   = (const float*)d_in[2];
    const float* W2   = (const float*)d_in[3];
    const float* b2   = (const float*)d_in[4];
    const float* W3   = (const float*)d_in[5];
    const float* b3   = (const float*)d_in[6];
    const float* Wlin = (const float*)d_in[7];
    const float* blin = (const float*)d_in[8];
    const int*   eidx = (const int*)d_in[9];
    const int*   batch= (const int*)d_in[10];

    const int N = in_sizes[0];            // 50000 (N_FEAT==1 so x is flat [N])
    const int E = in_sizes[9] / 2;        // 600000
    const int* src = eidx;
    const int* dst = eidx + E;

    // workspace carve-up (256B aligned)
    const size_t featB = (size_t)N * HIDDEN * sizeof(float);   // 25.6 MB
    char* ws = (char*)d_ws;
    float* bufX  = (float*)ws;                 ws += (featB + 255) & ~255ULL;  // pre
    float* bufY  = (float*)ws;                 ws += (featB + 255) & ~255ULL;  // agg / h
    float* dinv  = (float*)ws;                 ws += ((size_t)N * 4 + 255) & ~255ULL;
    float* sums  = (float*)ws;                 ws += ((size_t)N_GRAPHS * HIDDEN * 4 + 255) & ~255ULL;
    float* cnts  = (float*)ws;

    const int TB = 256;
    const int gN    = (N + TB - 1) / TB;
    const int gE    = (E + TB - 1) / TB;
    const int gFeat = (N * HIDDEN + TB - 1) / TB;      // 25000
    const int gScat = (int)(((long long)E * 32 + TB - 1) / TB);
    const int gPool = (int)(((long long)N * 32 + TB - 1) / TB);
    const int gGemm = N / 16;                          // 3125 (N multiple of 16)

    // degrees -> dinv
    zero_f32<<<gN, TB, 0, stream>>>(dinv, N);
    deg_kernel<<<gE, TB, 0, stream>>>(dst, dinv, E);
    dinv_kernel<<<gN, TB, 0, stream>>>(dinv, N);

    // ---- layer 1 (outer product) ----
    layer1_kernel<<<gFeat, TB, 0, stream>>>(x, W1, dinv, bufX, bufY, N);
    scatter_kernel<<<gScat, TB, 0, stream>>>(bufX, src, dst, dinv, bufY, E);
    finalize_kernel<<<gFeat, TB, 0, stream>>>(bufY, b1, N, 1);      // h1 in bufY

    // ---- layer 2 (WMMA fp32 GEMM) ----
    gemm_wmma_kernel<<<gGemm, TB, 0, stream>>>(bufY, W2, dinv, bufX, bufY);
    scatter_kernel<<<gScat, TB, 0, stream>>>(bufX, src, dst, dinv, bufY, E);
    finalize_kernel<<<gFeat, TB, 0, stream>>>(bufY, b2, N, 1);      // h2 in bufY

    // ---- layer 3 (WMMA fp32 GEMM, no relu) ----
    gemm_wmma_kernel<<<gGemm, TB, 0, stream>>>(bufY, W3, dinv, bufX, bufY);
    scatter_kernel<<<gScat, TB, 0, stream>>>(bufX, src, dst, dinv, bufY, E);
    finalize_kernel<<<gFeat, TB, 0, stream>>>(bufY, b3, N, 0);      // h3 in bufY

    // ---- global mean pool + linear head ----
    zero_f32<<<(N_GRAPHS * HIDDEN + N_GRAPHS + TB - 1) / TB, TB, 0, stream>>>(sums, N_GRAPHS * HIDDEN + N_GRAPHS); // sums then cnts are contiguous
    pool_kernel<<<gPool, TB, 0, stream>>>(bufY, batch, sums, cnts, N);
    head_kernel<<<(out_size + TB - 1) / TB, TB, 0, stream>>>(sums, cnts, Wlin, blin, (float*)d_out, out_size);
}